// CSTEncoder18_26963804685192
// MI455X (gfx1250) — compile-verified
//
#include <hip/hip_runtime.h>

// ---------------------------------------------------------------------------
// CST18 fit for gfx1250 (MI455X): one wave32 per batch element.
// G = [A|y]^T [A|y] accumulated with V_WMMA_F32_16X16X4_F32 (3 tiles, K=400).
// Ridge solve (d=18) + refit (d=17) via in-LDS Gaussian elimination.
// ---------------------------------------------------------------------------

typedef float v2f __attribute__((ext_vector_type(2)));
typedef float v8f __attribute__((ext_vector_type(8)));

#define NPTS   400
#define NBLK   13          // ceil(400/32)
#define RIDGE_F 1e-6f

struct WaveScratch {
  float rows[32][20];      // staged design rows: 19 cols (+pad)
  float G[19][20];         // Gram matrix [A|y]^T[A|y] (rows 0..18)
  float M[18][20];         // augmented system for elimination
  float sol1[20];
  float sol2[20];
};

// Gaussian elimination (no pivoting; SPD + ridge) on d x (d+1) system.
// M[i][j] = G[i][j] + ridge*delta_ij for j<d, M[i][d] = G[i][18] (rhs).
__device__ __forceinline__ void gauss_solve(volatile float (*G)[20],
                                            volatile float (*M)[20],
                                            volatile float* sol,
                                            int d, int lane) {
  for (int i = 0; i < d; ++i) {
    if (lane <= d) {
      float v = (lane == d) ? G[i][18] : G[i][lane];
      if (lane == i) v += RIDGE_F;
      M[i][lane] = v;
    }
  }
  __builtin_amdgcn_wave_barrier();
  // forward elimination: lane i owns row i
  for (int k = 0; k < d; ++k) {
    float piv = M[k][k];
    if (lane > k && lane < d) {
      float f = M[lane][k] / piv;
      for (int j = k; j <= d; ++j)
        M[lane][j] -= f * M[k][j];
    }
    __builtin_amdgcn_wave_barrier();
  }
  // back substitution (redundant across lanes; lane 0 publishes)
  for (int k = d - 1; k >= 0; --k) {
    float s = M[k][d];
    for (int j = k + 1; j < d; ++j)
      s -= M[k][j] * sol[j];
    if (lane == 0) sol[k] = s / M[k][k];
    __builtin_amdgcn_wave_barrier();
  }
}

__global__ __launch_bounds__(256) void cst18_kernel(const float* __restrict__ xg,
                                                    float* __restrict__ out,
                                                    int nbatch) {
  __shared__ WaveScratch wsm[8];
  const int lane = threadIdx.x & 31;
  const int wave = threadIdx.x >> 5;
  const int b    = blockIdx.x * 8 + wave;
  if (b >= nbatch) return;                      // uniform per wave

  WaveScratch& w = wsm[wave];
  const float2* xy = reinterpret_cast<const float2*>(xg) + (size_t)b * NPTS;

  // ---------------- pass 1: argmin(xc) (first index) + total sum(y) --------
  float runv = 3.0e38f; int runi = NPTS; float sumy = 0.f;
  for (int blk = 0; blk < NBLK; ++blk) {
    int n = blk * 32 + lane;
    float v = 3.0e38f;
    if (n < NPTS) {
      float2 p = xy[n];
      float xc = fminf(fmaxf(p.x, 0.f), 1.f);
      v = xc;
      sumy += p.y;
    }
    if (v < runv) { runv = v; runi = n; }       // strict < keeps earliest n
  }
  #pragma unroll
  for (int off = 16; off > 0; off >>= 1) {
    float ov = __shfl_xor(runv, off, 32);
    int   oi = __shfl_xor(runi, off, 32);
    float os = __shfl_xor(sumy, off, 32);
    sumy += os;
    if (ov < runv || (ov == runv && oi < runi)) { runv = ov; runi = oi; }
  }
  const int i_le    = runi;
  const int b_start = (i_le == NPTS - 1) ? (NPTS - 1) : (i_le + 1);
  const float cnt_a = (float)max(b_start, 1);
  const float cnt_b = (float)max(NPTS - b_start, 1);

  // ---------------- pass 1b: sum(y) over branch_a ---------------------------
  float suma = 0.f;
  for (int blk = 0; blk < NBLK; ++blk) {
    int n = blk * 32 + lane;
    if (n < b_start) { float2 p = xy[n]; suma += p.y; }
  }
  #pragma unroll
  for (int off = 16; off > 0; off >>= 1) suma += __shfl_xor(suma, off, 32);
  const float mean_a = suma / cnt_a;
  const float mean_b = (sumy - suma) / cnt_b;
  const bool upper_is_a = (mean_a >= mean_b);

  // ---------------- pass 2: build rows, accumulate G via WMMA f32 16x16x4 ---
  v8f c00 = {0,0,0,0,0,0,0,0};
  v8f c01 = {0,0,0,0,0,0,0,0};
  v8f c11 = {0,0,0,0,0,0,0,0};
  const float coef[8] = {1.f, 7.f, 21.f, 35.f, 35.f, 21.f, 7.f, 1.f};
  volatile float (*rows)[20] = (volatile float (*)[20])w.rows;

  const int cidx = lane & 15;                   // d-index within 16-wide tile
  const int rofs = (lane >> 4) * 2;             // K sub-offset per lane half

  for (int blk = 0; blk < NBLK; ++blk) {
    int n = blk * 32 + lane;
    // prefetch next block's points into cache
    if (blk + 1 < NBLK) {
      int n2 = n + 32;
      if (n2 < NPTS) __builtin_prefetch(&xy[n2], 0, 1);
    }
    float rowv[19];
    #pragma unroll
    for (int t = 0; t < 19; ++t) rowv[t] = 0.f;
    if (n < NPTS) {
      float2 p = xy[n];
      float xc  = fminf(fmaxf(p.x, 0.f), 1.f);
      float y   = p.y;
      float one = 1.f - xc;
      bool is_up = upper_is_a ? (n < b_start) : (n >= b_start);
      float pc = fminf(fmaxf(xc, 1e-8f), 1.f - 1e-8f);
      float C  = sqrtf(pc) * (1.f - pc);        // pc^0.5 * (1-pc)^1.0
      float pp[8], qq[8];
      float pa = 1.f, qa = 1.f;
      #pragma unroll
      for (int k = 0; k < 8; ++k) { pp[k] = pa; pa *= xc; qq[k] = qa; qa *= one; }
      #pragma unroll
      for (int k = 0; k < 8; ++k) {
        float cs = C * coef[k] * pp[k] * qq[7 - k];   // Bernstein * weight
        rowv[k]     = is_up ? 0.f : cs;               // lower block
        rowv[8 + k] = is_up ? cs  : 0.f;              // upper block
      }
      float one2 = one * one, one4 = one2 * one2, one8 = one4 * one4;
      rowv[16] = xc * one8 * sqrtf(one);              // LE col: x*(1-x)^8.5
      rowv[17] = is_up ? (0.5f * xc) : (-0.5f * xc);  // TE col
      rowv[18] = y;                                   // rhs column
    }
    #pragma unroll
    for (int t = 0; t < 19; ++t) rows[lane][t] = rowv[t];
    __builtin_amdgcn_wave_barrier();

    // 8 K-chunks of 4 rows each; identical registers serve SRC0 and SRC1
    #pragma unroll
    for (int kc = 0; kc < 8; ++kc) {
      int r = kc * 4 + rofs;
      v2f alo, ahi;
      alo.x = rows[r    ][cidx];
      alo.y = rows[r + 1][cidx];
      bool hv = cidx < 3;                         // cols 16..18 only
      ahi.x = hv ? rows[r    ][16 + cidx] : 0.f;
      ahi.y = hv ? rows[r + 1][16 + cidx] : 0.f;
      c00 = __builtin_amdgcn_wmma_f32_16x16x4_f32(false, alo, false, alo, (short)0, c00, false, false);
      c01 = __builtin_amdgcn_wmma_f32_16x16x4_f32(false, alo, false, ahi, (short)0, c01, false, false);
      c11 = __builtin_amdgcn_wmma_f32_16x16x4_f32(false, ahi, false, ahi, (short)0, c11, false, false);
    }
    __builtin_amdgcn_wave_barrier();
  }

  // ---------------- spill G tiles to LDS (C/D layout: N=lane%16, M=r+8*(lane/16))
  volatile float (*G)[20] = (volatile float (*)[20])w.G;
  const int mb = (lane >> 4) * 8;
  #pragma unroll
  for (int r2 = 0; r2 < 8; ++r2) G[mb + r2][cidx] = c00[r2];
  if (cidx < 3) {
    #pragma unroll
    for (int r2 = 0; r2 < 8; ++r2) G[mb + r2][16 + cidx] = c01[r2];
  }
  if (lane < 3) {                 // tile11: rows 16..18, cols 16..18
    G[16][16 + lane] = c11[0];
    G[17][16 + lane] = c11[1];
    G[18][16 + lane] = c11[2];
  }
  __builtin_amdgcn_wave_barrier();
  if (lane < 16) {                // mirror symmetric entries for rows 16,17
    G[16][lane] = G[lane][16];
    G[17][lane] = G[lane][17];
  }
  __builtin_amdgcn_wave_barrier();

  // ---------------- solves ---------------------------------------------------
  volatile float (*M)[20] = (volatile float (*)[20])w.M;
  gauss_solve(G, M, (volatile float*)w.sol1, 18, lane);
  gauss_solve(G, M, (volatile float*)w.sol2, 17, lane);
  __builtin_amdgcn_wave_barrier();

  if (lane < 18) {
    volatile float* s1 = (volatile float*)w.sol1;
    volatile float* s2 = (volatile float*)w.sol2;
    bool bad = s1[17] < 0.f;
    float val = bad ? ((lane < 17) ? s2[lane] : 0.f) : s1[lane];
    out[(size_t)b * 18 + lane] = val;
  }
}

extern "C" void kernel_launch(void* const* d_in, const int* in_sizes, int n_in,
                              void* d_out, int out_size, void* d_ws, size_t ws_size,
                              hipStream_t stream) {
  (void)n_in; (void)out_size; (void)d_ws; (void)ws_size;
  const float* x = (const float*)d_in[0];
  float* out = (float*)d_out;
  const int nbatch = in_sizes[0] / 800;         // 4096
  const int blocks = (nbatch + 7) / 8;          // 8 waves (batch elems) / block
  hipLaunchKernelGGL(cst18_kernel, dim3(blocks), dim3(256), 0, stream,
                     x, out, nbatch);
}